// RGCNLayer_45681272160381
// MI455X (gfx1250) — compile-verified
//
#include <hip/hip_runtime.h>
#include <hip/hip_bf16.h>

// ---------------- problem constants (from reference) ----------------
#define NN       100000
#define EE       1600000
#define IN_DIM   128
#define NHEAD    4
#define DDIM     64
#define HD       256           // NHEAD * DDIM
#define NEG_SLOPE 0.2f
#define K_APPNP  5
#define ALPHA    0.5f
#define BN_EPS   1e-5f

// ---------------- WMMA types ----------------
typedef float v2f __attribute__((ext_vector_type(2)));
typedef float v8f __attribute__((ext_vector_type(8)));

// order-preserving float <-> uint encoding for atomicMax on floats
__device__ __forceinline__ unsigned enc_f(float f) {
    unsigned u = __float_as_uint(f);
    return (u & 0x80000000u) ? ~u : (u | 0x80000000u);
}
__device__ __forceinline__ float dec_f(unsigned u) {
    return (u & 0x80000000u) ? __uint_as_float(u ^ 0x80000000u) : __uint_as_float(~u);
}

// =====================================================================
// fp32 WMMA GEMM:  out(M x OUTC) = A(M x 128) * B(128 x OUTC)
// Compile-time OUTC / TRANS / MODE so the K-loop is 32 straight-line
// v_wmma_f32_16x16x4_f32 with constant-offset addressing.
// Block = 128 threads = 4 waves. The block stages one 16x128 A tile in
// LDS (coalesced float4); each wave owns one 16x16 output column tile.
//   TRANS==0: B[k*OUTC+o]   TRANS==1: B[o*128+k]
//   MODE 0: out = C   MODE 1: out += C   MODE 2: out += relu(C + bias[o])
// =====================================================================
template <int OUTC, int TRANS, int MODE>
__global__ __launch_bounds__(128)
void gemm_wmma(const float* __restrict__ A, const float* __restrict__ W,
               float* __restrict__ out, const float* __restrict__ bias)
{
    __shared__ float As[16 * IN_DIM];           // 8 KB A tile, shared by 4 waves
    const int tile_m = blockIdx.x;
    const int wave   = threadIdx.x >> 5;
    const int lane   = threadIdx.x & 31;
    const int tile_n = blockIdx.y * 4 + wave;

    // cooperative coalesced float4 load of the 16x128 A tile
    {
        const float4* Ab = (const float4*)(A + (size_t)tile_m * 16 * IN_DIM);
        float4* AsV = (float4*)As;
        #pragma unroll
        for (int i = threadIdx.x; i < (16 * IN_DIM) / 4; i += 128) AsV[i] = Ab[i];
    }
    __syncthreads();

    const int r16  = lane & 15;       // A row / B,C column within tile
    const int kh   = lane >> 4;       // K-half select
    const int colg = tile_n * 16 + r16;

    v8f c = {0.f, 0.f, 0.f, 0.f, 0.f, 0.f, 0.f, 0.f};

    #pragma unroll
    for (int k0 = 0; k0 < IN_DIM; k0 += 4) {
        const int kb = k0 + 2 * kh;
        v2f a, b;
        // A 16x4 fp32 frag: lane holds M=r16, K = 2*kh + r   (from LDS)
        a.x = As[r16 * IN_DIM + kb + 0];
        a.y = As[r16 * IN_DIM + kb + 1];
        // B 4x16 fp32 frag: lane holds N=r16, K = 2*kh + r
        if (TRANS == 0) {
            b.x = W[(kb + 0) * OUTC + colg];
            b.y = W[(kb + 1) * OUTC + colg];
        } else {
            b.x = W[colg * IN_DIM + kb + 0];
            b.y = W[colg * IN_DIM + kb + 1];
        }
        c = __builtin_amdgcn_wmma_f32_16x16x4_f32(false, a, false, b,
                                                  (short)0, c, false, false);
    }

    // C frag layout: VGPR r -> row M = r + 8*kh, col N = r16
    const float bcol = (MODE == 2) ? bias[colg] : 0.f;
    #pragma unroll
    for (int r = 0; r < 8; ++r) {
        const int row = tile_m * 16 + r + 8 * kh;
        const size_t idx = (size_t)row * OUTC + colg;
        const float v = c[r];
        if (MODE == 0)      out[idx] = v;
        else if (MODE == 1) out[idx] += v;
        else                out[idx] += fmaxf(v + bcol, 0.f);
    }
}

// ---------------- attention projections el/er ----------------
__global__ void attn_proj(const float* __restrict__ feat,
                          const float* __restrict__ attn_l,
                          const float* __restrict__ attn_r,
                          float* __restrict__ el, float* __restrict__ er)
{
    int t = blockIdx.x * blockDim.x + threadIdx.x;
    if (t >= NN * NHEAD) return;
    const int n = t >> 2, h = t & 3;
    const float4* f4  = (const float4*)(feat + (size_t)n * HD + h * DDIM);
    const float4* al4 = (const float4*)(attn_l + h * DDIM);
    const float4* ar4 = (const float4*)(attn_r + h * DDIM);
    float sl = 0.f, sr = 0.f;
    #pragma unroll
    for (int i = 0; i < DDIM / 4; ++i) {
        float4 fv = f4[i], lv = al4[i], rv = ar4[i];
        sl += fv.x * lv.x + fv.y * lv.y + fv.z * lv.z + fv.w * lv.w;
        sr += fv.x * rv.x + fv.y * rv.y + fv.z * rv.z + fv.w * rv.w;
    }
    el[t] = sl; er[t] = sr;
}

// ---------------- edge softmax: pass 1 (max) ----------------
__global__ void edge_max(const int* __restrict__ src, const int* __restrict__ dst,
                         const float* __restrict__ el, const float* __restrict__ er,
                         unsigned* __restrict__ emax_enc)
{
    int e = blockIdx.x * blockDim.x + threadIdx.x;
    if (e >= EE) return;
    const int s = src[e], d = dst[e];
    #pragma unroll
    for (int h = 0; h < NHEAD; ++h) {
        float x = el[s * NHEAD + h] + er[d * NHEAD + h];
        x = (x > 0.f) ? x : NEG_SLOPE * x;
        atomicMax(&emax_enc[d * NHEAD + h], enc_f(x));
    }
}

// ---------------- edge softmax: pass 2 (denominator) + degree ----------------
__global__ void edge_denom(const int* __restrict__ src, const int* __restrict__ dst,
                           const float* __restrict__ el, const float* __restrict__ er,
                           const unsigned* __restrict__ emax_enc,
                           float* __restrict__ denom, float* __restrict__ deg)
{
    int e = blockIdx.x * blockDim.x + threadIdx.x;
    if (e >= EE) return;
    const int s = src[e], d = dst[e];
    #pragma unroll
    for (int h = 0; h < NHEAD; ++h) {
        float x = el[s * NHEAD + h] + er[d * NHEAD + h];
        x = (x > 0.f) ? x : NEG_SLOPE * x;
        float em = dec_f(emax_enc[d * NHEAD + h]);
        atomicAdd(&denom[d * NHEAD + h], __expf(x - em));
    }
    atomicAdd(&deg[d], 1.f);
}

// ---------------- GAT message aggregation (64 threads / edge) ----------------
__global__ void edge_message(const int* __restrict__ src, const int* __restrict__ dst,
                             const float* __restrict__ el, const float* __restrict__ er,
                             const unsigned* __restrict__ emax_enc,
                             const float* __restrict__ denom,
                             const float* __restrict__ feat,
                             float* __restrict__ agg)
{
    int t = blockIdx.x * blockDim.x + threadIdx.x;
    if (t >= EE * 64) return;
    const int e = t >> 6;
    const int j = t & 63;
    const int h = j >> 4;             // head
    const int d0 = (j & 15) * 4;      // float4 within D
    const int s = src[e], d = dst[e];

    float x = el[s * NHEAD + h] + er[d * NHEAD + h];
    x = (x > 0.f) ? x : NEG_SLOPE * x;
    const float em  = dec_f(emax_enc[d * NHEAD + h]);
    const float den = fmaxf(denom[d * NHEAD + h], 1e-9f);
    const float a   = __expf(x - em) / den;

    const float4 fv = *(const float4*)(feat + (size_t)s * HD + h * DDIM + d0);
    float* p = agg + (size_t)d * HD + h * DDIM + d0;
    atomicAdd(p + 0, fv.x * a);
    atomicAdd(p + 1, fv.y * a);
    atomicAdd(p + 2, fv.z * a);
    atomicAdd(p + 3, fv.w * a);
}

// ---------------- relu(agg+res+bias) then Conv1d head-reduce ----------------
__global__ void conv_reduce(const float* __restrict__ agg,
                            const float* __restrict__ gat_bias,
                            const float* __restrict__ conv_w,
                            const float* __restrict__ conv_b,
                            float* __restrict__ h, float* __restrict__ h0)
{
    int t = blockIdx.x * blockDim.x + threadIdx.x;
    if (t >= NN * DDIM) return;
    const int n = t >> 6, d = t & 63;
    float acc = conv_b[0];
    #pragma unroll
    for (int hh = 0; hh < NHEAD; ++hh) {
        float v = agg[(size_t)n * HD + hh * DDIM + d] + gat_bias[hh * DDIM + d];
        acc += conv_w[hh] * fmaxf(v, 0.f);
    }
    h[t] = acc; h0[t] = acc;
}

// ---------------- APPNP helpers ----------------
__global__ void deg_norm(const float* __restrict__ deg, float* __restrict__ norm)
{
    int n = blockIdx.x * blockDim.x + threadIdx.x;
    if (n >= NN) return;
    norm[n] = rsqrtf(fmaxf(deg[n], 1.f));
}

__global__ void appnp_scale_zero(const float* __restrict__ h, const float* __restrict__ norm,
                                 float* __restrict__ tmp, float* __restrict__ hnew)
{
    int t = blockIdx.x * blockDim.x + threadIdx.x;
    if (t >= NN * DDIM) return;
    tmp[t]  = h[t] * norm[t >> 6];
    hnew[t] = 0.f;
}

__global__ void appnp_prop(const int* __restrict__ src, const int* __restrict__ dst,
                           const float* __restrict__ tmp, float* __restrict__ hnew)
{
    int t = blockIdx.x * blockDim.x + threadIdx.x;
    if (t >= EE * 16) return;
    const int e = t >> 4;
    const int d0 = (t & 15) * 4;
    const int s = src[e], d = dst[e];
    const float4 v = *(const float4*)(tmp + (size_t)s * DDIM + d0);
    float* p = hnew + (size_t)d * DDIM + d0;
    atomicAdd(p + 0, v.x); atomicAdd(p + 1, v.y);
    atomicAdd(p + 2, v.z); atomicAdd(p + 3, v.w);
}

__global__ void appnp_update(const float* __restrict__ hnew, const float* __restrict__ norm,
                             const float* __restrict__ h0, float* __restrict__ h)
{
    int t = blockIdx.x * blockDim.x + threadIdx.x;
    if (t >= NN * DDIM) return;
    h[t] = (1.f - ALPHA) * hnew[t] * norm[t >> 6] + ALPHA * h0[t];
}

// ---------------- BatchNorm ----------------
__global__ void bn_stats(const float* __restrict__ h, float* __restrict__ stats)
{
    __shared__ float ssum[256], ssq[256];
    const int t = threadIdx.x;
    const int col = t & 63, rg = t >> 6;
    float s = 0.f, q = 0.f;
    for (int r = blockIdx.x * 4 + rg; r < NN; r += gridDim.x * 4) {
        float v = h[(size_t)r * DDIM + col];
        s += v; q += v * v;
    }
    ssum[t] = s; ssq[t] = q;
    __syncthreads();
    if (t < 64) {
        s = ssum[t] + ssum[t + 64] + ssum[t + 128] + ssum[t + 192];
        q = ssq[t]  + ssq[t + 64]  + ssq[t + 128]  + ssq[t + 192];
        atomicAdd(&stats[t], s);
        atomicAdd(&stats[64 + t], q);
    }
}

__global__ void bn_apply(const float* __restrict__ h, const float* __restrict__ stats,
                         const float* __restrict__ gamma, const float* __restrict__ beta,
                         float* __restrict__ out)
{
    int t = blockIdx.x * blockDim.x + threadIdx.x;
    if (t >= NN * DDIM) return;
    const int d = t & 63;
    const float inv_n = 1.f / (float)NN;
    const float mu  = stats[d] * inv_n;
    const float var = stats[64 + d] * inv_n - mu * mu;
    out[t] = (h[t] - mu) * rsqrtf(var + BN_EPS) * gamma[d] + beta[d];
}

// =====================================================================
extern "C" void kernel_launch(void* const* d_in, const int* in_sizes, int n_in,
                              void* d_out, int out_size, void* d_ws, size_t ws_size,
                              hipStream_t stream)
{
    (void)in_sizes; (void)n_in; (void)out_size; (void)ws_size;

    const float* node_feats = (const float*)d_in[0];
    const int*   src        = (const int*)  d_in[1];
    const int*   dst        = (const int*)  d_in[2];
    const float* W_fc       = (const float*)d_in[3];
    const float* attn_l     = (const float*)d_in[4];
    const float* attn_r     = (const float*)d_in[5];
    const float* W_res_gat  = (const float*)d_in[6];
    const float* gat_bias   = (const float*)d_in[7];
    const float* conv_w     = (const float*)d_in[8];
    const float* conv_b     = (const float*)d_in[9];
    const float* W_res      = (const float*)d_in[10];
    const float* b_res      = (const float*)d_in[11];
    const float* gamma      = (const float*)d_in[12];
    const float* beta       = (const float*)d_in[13];
    float* out = (float*)d_out;

    // ---- workspace layout (floats) ----
    float* ws = (float*)d_ws;
    size_t o = 0;
    float*    feat     = ws + o;               o += (size_t)NN * HD;     // 25.6M
    float*    agg      = ws + o;               o += (size_t)NN * HD;     // 25.6M
    float*    el       = ws + o;               o += (size_t)NN * NHEAD;
    float*    er       = ws + o;               o += (size_t)NN * NHEAD;
    unsigned* emax_enc = (unsigned*)(ws + o);  o += (size_t)NN * NHEAD;
    float*    denom    = ws + o;               o += (size_t)NN * NHEAD;
    float*    deg      = ws + o;               o += (size_t)NN;
    float*    nrm      = ws + o;               o += (size_t)NN;
    float*    h        = ws + o;               o += (size_t)NN * DDIM;
    float*    h0       = ws + o;               o += (size_t)NN * DDIM;
    float*    tmp      = ws + o;               o += (size_t)NN * DDIM;
    float*    hnew     = ws + o;               o += (size_t)NN * DDIM;
    float*    stats    = ws + o;               o += 128;

    // ---- zero-init accumulators (0x0 == -inf sentinel for emax encoding) ----
    hipMemsetAsync(agg,      0, (size_t)NN * HD    * sizeof(float), stream);
    hipMemsetAsync(emax_enc, 0, (size_t)NN * NHEAD * sizeof(unsigned), stream);
    hipMemsetAsync(denom,    0, (size_t)NN * NHEAD * sizeof(float), stream);
    hipMemsetAsync(deg,      0, (size_t)NN         * sizeof(float), stream);
    hipMemsetAsync(stats,    0, 128 * sizeof(float), stream);

    const int MT = NN / 16;  // 6250 row tiles (exact)

    // ---- GAT: feat = node_feats @ W_fc (WMMA fp32) ----
    gemm_wmma<HD, 0, 0><<<dim3(MT, HD / 64), 128, 0, stream>>>(node_feats, W_fc, feat, nullptr);

    // ---- el / er projections ----
    attn_proj<<<(NN * NHEAD + 255) / 256, 256, 0, stream>>>(feat, attn_l, attn_r, el, er);

    // ---- edge softmax ----
    edge_max  <<<(EE + 255) / 256, 256, 0, stream>>>(src, dst, el, er, emax_enc);
    edge_denom<<<(EE + 255) / 256, 256, 0, stream>>>(src, dst, el, er, emax_enc, denom, deg);

    // ---- message aggregation ----
    edge_message<<<(EE * 64 + 255) / 256, 256, 0, stream>>>(src, dst, el, er, emax_enc,
                                                            denom, feat, agg);

    // ---- residual projection: agg += node_feats @ W_res_gat (WMMA fp32) ----
    gemm_wmma<HD, 0, 1><<<dim3(MT, HD / 64), 128, 0, stream>>>(node_feats, W_res_gat, agg, nullptr);

    // ---- relu + bias + conv head-reduce -> h, h0 ----
    conv_reduce<<<(NN * DDIM + 255) / 256, 256, 0, stream>>>(agg, gat_bias, conv_w, conv_b, h, h0);

    // ---- APPNP ----
    deg_norm<<<(NN + 255) / 256, 256, 0, stream>>>(deg, nrm);
    for (int k = 0; k < K_APPNP; ++k) {
        appnp_scale_zero<<<(NN * DDIM + 255) / 256, 256, 0, stream>>>(h, nrm, tmp, hnew);
        appnp_prop      <<<(EE * 16 + 255) / 256, 256, 0, stream>>>(src, dst, tmp, hnew);
        appnp_update    <<<(NN * DDIM + 255) / 256, 256, 0, stream>>>(hnew, nrm, h0, h);
    }

    // ---- h += relu(node_feats @ W_res.T + b_res)  (WMMA fp32, B transposed) ----
    gemm_wmma<DDIM, 1, 2><<<dim3(MT, 1), 128, 0, stream>>>(node_feats, W_res, h, b_res);

    // ---- BatchNorm ----
    bn_stats<<<256, 256, 0, stream>>>(h, stats);
    bn_apply<<<(NN * DDIM + 255) / 256, 256, 0, stream>>>(h, stats, gamma, beta, out);
}